// GCN_38766374814041
// MI455X (gfx1250) — compile-verified
//
#include <hip/hip_runtime.h>
#include <hip/hip_bf16.h>

#define NN 100000
#define EE 1600000
#define GENE_DIM 39
#define NUM_CLASSES 5
#define EPSF 1e-5f

typedef __attribute__((ext_vector_type(2))) float v2f;
typedef __attribute__((ext_vector_type(8))) float v8f;

// ---------------------------------------------------------------------------
// utility: zero a region of 32-bit words (grid-stride)
// ---------------------------------------------------------------------------
__global__ __launch_bounds__(256) void zero_words(unsigned* __restrict__ p, long n) {
    long i = (long)blockIdx.x * blockDim.x + threadIdx.x;
    long stride = (long)gridDim.x * blockDim.x;
    for (; i < n; i += stride) p[i] = 0u;
}

// ---------------------------------------------------------------------------
// degree count (in-degree, self loop added analytically later)
// ---------------------------------------------------------------------------
__global__ __launch_bounds__(256) void degree_kernel(const long long* __restrict__ dst,
                                                     int* __restrict__ degcnt) {
    int e = blockIdx.x * blockDim.x + threadIdx.x;
    if (e < EE) atomicAdd(&degcnt[(int)dst[e]], 1);
}

// ---------------------------------------------------------------------------
// single-block exclusive scan of degcnt -> row_ptr[N+1]
// ---------------------------------------------------------------------------
__global__ __launch_bounds__(1024) void scan_kernel(const int* __restrict__ degcnt,
                                                    int* __restrict__ row_ptr) {
    __shared__ int ls[1024];
    const int t = threadIdx.x;
    const int CH = (NN + 1023) / 1024;
    int base = t * CH;
    int s = 0;
    for (int j = 0; j < CH; ++j) {
        int i = base + j;
        if (i < NN) s += degcnt[i];
    }
    ls[t] = s;
    __syncthreads();
    for (int off = 1; off < 1024; off <<= 1) {
        int add = (t >= off) ? ls[t - off] : 0;
        __syncthreads();
        ls[t] += add;
        __syncthreads();
    }
    int run = (t > 0) ? ls[t - 1] : 0;   // exclusive prefix of this chunk
    for (int j = 0; j < CH; ++j) {
        int i = base + j;
        if (i < NN) { row_ptr[i] = run; run += degcnt[i]; }
    }
    if (t == 1023) row_ptr[NN] = ls[1023];
}

// ---------------------------------------------------------------------------
// dinv = rsqrt(deg+1) ; cursor = row_ptr copy
// ---------------------------------------------------------------------------
__global__ __launch_bounds__(256) void dinv_cursor_kernel(const int* __restrict__ degcnt,
                                                          const int* __restrict__ row_ptr,
                                                          float* __restrict__ dinv,
                                                          int* __restrict__ cursor) {
    int i = blockIdx.x * blockDim.x + threadIdx.x;
    if (i < NN) {
        dinv[i] = rsqrtf((float)(degcnt[i] + 1));  // +1 self loop; always > 0
        cursor[i] = row_ptr[i];
    }
}

// ---------------------------------------------------------------------------
// fill CSR (sorted-by-dst) with src index and precomputed GCN edge weight
// ---------------------------------------------------------------------------
__global__ __launch_bounds__(256) void csr_fill_kernel(const long long* __restrict__ src,
                                                       const long long* __restrict__ dst,
                                                       const float* __restrict__ dinv,
                                                       int* __restrict__ cursor,
                                                       int* __restrict__ col,
                                                       float* __restrict__ ew) {
    int e = blockIdx.x * blockDim.x + threadIdx.x;
    if (e < EE) {
        int s = (int)src[e];
        int d = (int)dst[e];
        int pos = atomicAdd(&cursor[d], 1);
        col[pos] = s;
        ew[pos]  = dinv[s] * dinv[d];
    }
}

// ---------------------------------------------------------------------------
// fp32 WMMA GEMM: Y[N x DOUT] = H[N x DIN] @ W[DIN x DOUT]
// one wave per 16x16 output tile, K stepped by 4 (V_WMMA_F32_16X16X4_F32)
// N = 100000 = 6250 * 16 -> no M masking. Only DIN=39 needs K padding.
// ---------------------------------------------------------------------------
template <int DIN, int DOUT>
__global__ __launch_bounds__(256) void gemm_wmma(const float* __restrict__ H,
                                                 const float* __restrict__ W,
                                                 float* __restrict__ Y) {
    constexpr int TN = DOUT / 16;
    constexpr int MT = NN / 16;            // 6250
    int wave = (blockIdx.x * 256 + threadIdx.x) >> 5;
    if (wave >= MT * TN) return;           // wave-uniform: EXEC stays full below
    int lane = threadIdx.x & 31;
    int tm = wave / TN, tn = wave % TN;
    int row0 = tm * 16, col0 = tn * 16;
    int l16  = lane & 15;
    int khalf = (lane >> 4) << 1;          // lanes 0-15 -> K+0/1, lanes 16-31 -> K+2/3

    v8f c = {};
    constexpr int KT = (DIN + 3) / 4;
    const float* hrow = H + (long)(row0 + l16) * DIN;
    for (int kt = 0; kt < KT; ++kt) {
        int ka = kt * 4 + khalf;
        v2f a, b;
        if constexpr (DIN % 4 == 0) {
            a.x = hrow[ka];
            a.y = hrow[ka + 1];
            b.x = W[(long)ka * DOUT + col0 + l16];
            b.y = W[(long)(ka + 1) * DOUT + col0 + l16];
        } else {
            a.x = (ka     < DIN) ? hrow[ka]     : 0.0f;
            a.y = (ka + 1 < DIN) ? hrow[ka + 1] : 0.0f;
            b.x = (ka     < DIN) ? W[(long)ka * DOUT + col0 + l16]       : 0.0f;
            b.y = (ka + 1 < DIN) ? W[(long)(ka + 1) * DOUT + col0 + l16] : 0.0f;
        }
        c = __builtin_amdgcn_wmma_f32_16x16x4_f32(false, a, false, b,
                                                  (short)0, c, false, false);
    }
    // C/D layout: VGPR r -> M=r (lanes 0-15) / M=r+8 (lanes 16-31), N=lane%16
    int crow = row0 + ((lane >= 16) ? 8 : 0);
    int ccol = col0 + l16;
#pragma unroll
    for (int r = 0; r < 8; ++r)
        Y[(long)(crow + r) * DOUT + ccol] = c[r];
}

// ---------------------------------------------------------------------------
// gather-side aggregation (no feature atomics): one block per node,
// one thread per output feature; (col,w) staged through LDS in chunks.
// out = relu( sum_j w_ij * y_j  +  dinv_i^2 * y_i  +  b )
// ---------------------------------------------------------------------------
template <int DOUT>
__global__ __launch_bounds__(DOUT) void aggregate_kernel(const float* __restrict__ Y,
                                                         const int* __restrict__ rp,
                                                         const int* __restrict__ col,
                                                         const float* __restrict__ ew,
                                                         const float* __restrict__ dinv,
                                                         const float* __restrict__ bias,
                                                         float* __restrict__ Hout) {
    __shared__ int   scol[DOUT];
    __shared__ float sw[DOUT];
    int i = blockIdx.x;
    int f = threadIdx.x;
    int e0 = rp[i], e1 = rp[i + 1];
    float di = dinv[i];
    float acc = Y[(long)i * DOUT + f] * di * di;   // self loop
    for (int eb = e0; eb < e1; eb += DOUT) {
        int cnt = min(DOUT, e1 - eb);
        __syncthreads();
        if (f < cnt) { scol[f] = col[eb + f]; sw[f] = ew[eb + f]; }
        __syncthreads();
        for (int j = 0; j < cnt; ++j)
            acc += sw[j] * Y[(long)scol[j] * DOUT + f];   // coalesced across f
    }
    acc += bias[f];
    Hout[(long)i * DOUT + f] = fmaxf(acc, 0.0f);
}

// ---------------------------------------------------------------------------
// BatchNorm stats: per-column sum & sum-of-squares via LDS tree + few atomics
// ---------------------------------------------------------------------------
template <int DOUT>
__global__ __launch_bounds__(256) void bn_stats_kernel(const float* __restrict__ H,
                                                       float* __restrict__ sum,
                                                       float* __restrict__ sq) {
    constexpr int T = 256;
    constexpr int R = T / DOUT;   // node-rows handled per block per step
    __shared__ float ls[T], lq[T];
    int f = threadIdx.x % DOUT;
    int r = threadIdx.x / DOUT;
    float s = 0.0f, q = 0.0f;
    for (int i = blockIdx.x * R + r; i < NN; i += gridDim.x * R) {
        float v = H[(long)i * DOUT + f];
        s += v; q += v * v;
    }
    ls[threadIdx.x] = s; lq[threadIdx.x] = q;
    __syncthreads();
    for (int st = T / 2; st >= DOUT; st >>= 1) {
        if (threadIdx.x < st) {
            ls[threadIdx.x] += ls[threadIdx.x + st];
            lq[threadIdx.x] += lq[threadIdx.x + st];
        }
        __syncthreads();
    }
    if (threadIdx.x < DOUT) {
        atomicAdd(&sum[f], ls[threadIdx.x]);
        atomicAdd(&sq[f],  lq[threadIdx.x]);
    }
}

// ---------------------------------------------------------------------------
// BatchNorm apply (in place): g*(h-mu)*rsqrt(var+eps)+be, biased variance
// ---------------------------------------------------------------------------
template <int DOUT>
__global__ __launch_bounds__(256) void bn_apply_kernel(float* __restrict__ H,
                                                       const float* __restrict__ sum,
                                                       const float* __restrict__ sq,
                                                       const float* __restrict__ g,
                                                       const float* __restrict__ be) {
    long idx = (long)blockIdx.x * blockDim.x + threadIdx.x;
    long total = (long)NN * DOUT;
    long stride = (long)gridDim.x * blockDim.x;
    const float invn = 1.0f / (float)NN;
    for (; idx < total; idx += stride) {
        int f = (int)(idx % DOUT);
        float mu  = sum[f] * invn;
        float var = sq[f] * invn - mu * mu;
        float v = H[idx];
        H[idx] = g[f] * (v - mu) * rsqrtf(var + EPSF) + be[f];
    }
}

// ---------------------------------------------------------------------------
// classifier: out = h @ Wc + bc (32 -> 5) and copy h into output slot 2
// ---------------------------------------------------------------------------
__global__ __launch_bounds__(256) void classifier_kernel(const float* __restrict__ H,
                                                         const float* __restrict__ Wc,
                                                         const float* __restrict__ bc,
                                                         float* __restrict__ out,
                                                         float* __restrict__ hcopy) {
    int i = blockIdx.x * blockDim.x + threadIdx.x;
    if (i >= NN) return;
    float hrow[32];
#pragma unroll
    for (int k = 0; k < 32; ++k) {
        float v = H[(long)i * 32 + k];
        hrow[k] = v;
        hcopy[(long)i * 32 + k] = v;
    }
#pragma unroll
    for (int c = 0; c < NUM_CLASSES; ++c) {
        float acc = bc[c];
#pragma unroll
        for (int k = 0; k < 32; ++k)
            acc += hrow[k] * Wc[k * NUM_CLASSES + c];
        out[(long)i * NUM_CLASSES + c] = acc;
    }
}

// ---------------------------------------------------------------------------
// launcher
// ---------------------------------------------------------------------------
static inline size_t align256(size_t x) { return (x + 255) & ~(size_t)255; }

extern "C" void kernel_launch(void* const* d_in, const int* in_sizes, int n_in,
                              void* d_out, int out_size, void* d_ws, size_t ws_size,
                              hipStream_t stream) {
    const float*     x    = (const float*)d_in[0];
    const long long* ei   = (const long long*)d_in[1];   // int64 [2, E] flat
    const long long* src64 = ei;
    const long long* dst64 = ei + EE;
    const float* W1 = (const float*)d_in[2],  *b1 = (const float*)d_in[3];
    const float* g1 = (const float*)d_in[4],  *be1 = (const float*)d_in[5];
    const float* W2 = (const float*)d_in[6],  *b2 = (const float*)d_in[7];
    const float* g2 = (const float*)d_in[8],  *be2 = (const float*)d_in[9];
    const float* W3 = (const float*)d_in[10], *b3 = (const float*)d_in[11];
    const float* g3 = (const float*)d_in[12], *be3 = (const float*)d_in[13];
    const float* W4 = (const float*)d_in[14], *b4 = (const float*)d_in[15];
    const float* g4 = (const float*)d_in[16], *be4 = (const float*)d_in[17];
    const float* Wc = (const float*)d_in[18], *bc = (const float*)d_in[19];

    // -------- workspace carve-up (~117 MB; everything feature-sized lives in L2)
    char* ws = (char*)d_ws;
    size_t off = 0;
    int*   degcnt = (int*)(ws + off);   off = align256(off + (size_t)NN * 4);
    int*   rowptr = (int*)(ws + off);   off = align256(off + (size_t)(NN + 1) * 4);
    int*   cursor = (int*)(ws + off);   off = align256(off + (size_t)NN * 4);
    float* dinv   = (float*)(ws + off); off = align256(off + (size_t)NN * 4);
    int*   col    = (int*)(ws + off);   off = align256(off + (size_t)EE * 4);
    float* ew     = (float*)(ws + off); off = align256(off + (size_t)EE * 4);
    float* bufA   = (float*)(ws + off); off = align256(off + (size_t)NN * 128 * 4);
    float* bufB   = (float*)(ws + off); off = align256(off + (size_t)NN * 128 * 4);
    float* bnsum  = (float*)(ws + off); off = align256(off + 128 * 4);
    float* bnsq   = (float*)(ws + off); off = align256(off + 128 * 4);
    (void)ws_size; (void)in_sizes; (void)n_in; (void)out_size;

    float* out_logits = (float*)d_out;
    float* out_hidden = (float*)d_out + (size_t)NN * NUM_CLASSES;

    const int EB = (EE + 255) / 256;   // 6250
    const int NB = (NN + 255) / 256;   // 391

    // -------- graph preprocessing (once per launch) --------
    zero_words<<<1024, 256, 0, stream>>>((unsigned*)degcnt, NN);
    degree_kernel<<<EB, 256, 0, stream>>>(dst64, degcnt);
    scan_kernel<<<1, 1024, 0, stream>>>(degcnt, rowptr);
    dinv_cursor_kernel<<<NB, 256, 0, stream>>>(degcnt, rowptr, dinv, cursor);
    csr_fill_kernel<<<EB, 256, 0, stream>>>(src64, dst64, dinv, cursor, col, ew);

    // -------- layer 1: 39 -> 64 --------
    {
        constexpr int DIN = GENE_DIM, DOUT = 64;
        int waves = (NN / 16) * (DOUT / 16);
        gemm_wmma<DIN, DOUT><<<(waves + 7) / 8, 256, 0, stream>>>(x, W1, bufA);
        aggregate_kernel<DOUT><<<NN, DOUT, 0, stream>>>(bufA, rowptr, col, ew, dinv, b1, bufB);
        zero_words<<<1, 256, 0, stream>>>((unsigned*)bnsum, 256);
        bn_stats_kernel<DOUT><<<256, 256, 0, stream>>>(bufB, bnsum, bnsq);
        bn_apply_kernel<DOUT><<<1024, 256, 0, stream>>>(bufB, bnsum, bnsq, g1, be1);
    }
    // -------- layer 2: 64 -> 128 --------
    {
        constexpr int DIN = 64, DOUT = 128;
        int waves = (NN / 16) * (DOUT / 16);
        gemm_wmma<DIN, DOUT><<<(waves + 7) / 8, 256, 0, stream>>>(bufB, W2, bufA);
        aggregate_kernel<DOUT><<<NN, DOUT, 0, stream>>>(bufA, rowptr, col, ew, dinv, b2, bufB);
        zero_words<<<1, 256, 0, stream>>>((unsigned*)bnsum, 256);
        bn_stats_kernel<DOUT><<<256, 256, 0, stream>>>(bufB, bnsum, bnsq);
        bn_apply_kernel<DOUT><<<1024, 256, 0, stream>>>(bufB, bnsum, bnsq, g2, be2);
    }
    // -------- layer 3: 128 -> 64 --------
    {
        constexpr int DIN = 128, DOUT = 64;
        int waves = (NN / 16) * (DOUT / 16);
        gemm_wmma<DIN, DOUT><<<(waves + 7) / 8, 256, 0, stream>>>(bufB, W3, bufA);
        aggregate_kernel<DOUT><<<NN, DOUT, 0, stream>>>(bufA, rowptr, col, ew, dinv, b3, bufB);
        zero_words<<<1, 256, 0, stream>>>((unsigned*)bnsum, 256);
        bn_stats_kernel<DOUT><<<256, 256, 0, stream>>>(bufB, bnsum, bnsq);
        bn_apply_kernel<DOUT><<<1024, 256, 0, stream>>>(bufB, bnsum, bnsq, g3, be3);
    }
    // -------- layer 4: 64 -> 32 --------
    {
        constexpr int DIN = 64, DOUT = 32;
        int waves = (NN / 16) * (DOUT / 16);
        gemm_wmma<DIN, DOUT><<<(waves + 7) / 8, 256, 0, stream>>>(bufB, W4, bufA);
        aggregate_kernel<DOUT><<<NN, DOUT, 0, stream>>>(bufA, rowptr, col, ew, dinv, b4, bufB);
        zero_words<<<1, 256, 0, stream>>>((unsigned*)bnsum, 256);
        bn_stats_kernel<DOUT><<<256, 256, 0, stream>>>(bufB, bnsum, bnsq);
        bn_apply_kernel<DOUT><<<1024, 256, 0, stream>>>(bufB, bnsum, bnsq, g4, be4);
    }
    // -------- classifier + hidden copy --------
    classifier_kernel<<<NB, 256, 0, stream>>>(bufB, Wc, bc, out_logits, out_hidden);
}